// DiscriminativeEmbeddingLoss_90400471646631
// MI455X (gfx1250) — compile-verified
//
#include <hip/hip_runtime.h>

typedef __attribute__((ext_vector_type(2))) float v2f;
typedef __attribute__((ext_vector_type(8))) float v8f;

// Problem constants (match reference setup_inputs)
constexpr int BB = 8, DD = 32, HH = 512, WW = 512, KK = 16;
constexpr int NN = HH * WW;            // pixels per image
constexpr int IGN = 255;
constexpr float DVAR = 0.5f, DDIST = 1.5f;
constexpr float PULLW = 1.0f, PUSHW = 1.0f, REGW = 0.001f;

// Workspace layout (in floats)
constexpr int OFF_COUNTS  = 0;                           // [B][K]
constexpr int OFF_SUMS    = OFF_COUNTS + BB * KK;        // [B][K][D]
constexpr int OFF_CENTERS = OFF_SUMS + BB * KK * DD;     // [B][K][D]
constexpr int OFF_CNORM2  = OFF_CENTERS + BB * KK * DD;  // [B][K]
constexpr int OFF_VALID   = OFF_CNORM2 + BB * KK;        // [B][K]
constexpr int OFF_PULL    = OFF_VALID + BB * KK;         // [B][K]
constexpr int OFF_IMG     = OFF_PULL + BB * KK;          // [B][4] push,reg,nids,vb
constexpr int WS_FLOATS   = OFF_IMG + BB * 4;

constexpr int BLOCKS_PER_IMG = 64;
constexpr int PIX_PER_BLOCK  = NN / BLOCKS_PER_IMG;      // 4096

__global__ void k_zero(float* __restrict__ ws) {
  int i = blockIdx.x * blockDim.x + threadIdx.x;
  if (i < WS_FLOATS) ws[i] = 0.f;
}

// ---------------- Pass 1: per-instance counts & embedding sums via WMMA ----------------
// sums[k][d] = sum_p onehot[k][p] * e[p][d]  as 16x16x4 f32 WMMAs, K-dim = 4 pixels.
// A = one-hot (register-built from seg compares), B = embeddings transposed through LDS.
__global__ __launch_bounds__(256) void k_pass1(const float* __restrict__ emb,
                                               const int* __restrict__ inst,
                                               const unsigned char* __restrict__ vmask,
                                               float* __restrict__ ws) {
  __shared__ float sE[8][16 * 36];   // per-wave staging tile [pixel][dim], stride 36 (8B aligned pairs)
  __shared__ int   sSeg[8][16];      // per-wave masked segment ids (-1 = invalid)
  __shared__ float s_sum[KK * 33];   // block-level partial sums
  __shared__ float s_cnt[KK];
  const int tid = threadIdx.x;
  for (int i = tid; i < KK * 33; i += 256) s_sum[i] = 0.f;
  if (tid < KK) s_cnt[tid] = 0.f;
  __syncthreads();

  const int b    = blockIdx.x / BLOCKS_PER_IMG;
  const int blk  = blockIdx.x % BLOCKS_PER_IMG;
  const int wave = tid >> 5;
  const int l    = tid & 31;
  const int row  = l & 15;
  const float* e0 = emb + (size_t)b * DD * NN;
  const int wbase = blk * PIX_PER_BLOCK + wave * 512;   // 512 pixels per wave

  v8f accLo = {}, accHi = {};        // persistent sums tiles: dims 0-15 / 16-31
  float* myE  = &sE[wave][0];
  int*   mySeg = &sSeg[wave][0];

  for (int t = 0; t < 32; ++t) {
    const int p0 = wbase + t * 16;
    // coalesced global loads (pixel-contiguous per d-plane) -> LDS transpose staging
#pragma unroll
    for (int c = 0; c < 8; ++c) {
      const int d0 = (l < 16) ? (4 * c) : (4 * c + 2);
      const float x = e0[(size_t)d0 * NN + p0 + row];
      const float y = e0[(size_t)(d0 + 1) * NN + p0 + row];
      myE[row * 36 + d0]     = x;
      myE[row * 36 + d0 + 1] = y;
    }
    // streaming prefetch a few tiles ahead (global_prefetch_b8)
    __builtin_prefetch(&e0[(size_t)((l < 16) ? 0 : 2) * NN + p0 + 64 + row], 0, 0);
    if (l < 16) {
      const int p = p0 + row;
      const int s = inst[(size_t)b * NN + p];
      const bool msk = vmask[(size_t)b * NN + p] && (s != IGN) && (s >= 0) && (s < KK);
      mySeg[row] = msk ? s : -1;
      if (msk) atomicAdd(&s_cnt[s], 1.f);
    }
    __syncthreads();
#pragma unroll
    for (int g = 0; g < 4; ++g) {    // 4 pixel-groups of 4 (WMMA K-dim)
      const int j0 = (l < 16) ? 0 : 2;   // A layout: lanes0-15 hold K=0,1; lanes16-31 hold K=2,3
      const int sa = mySeg[g * 4 + j0];
      const int sb = mySeg[g * 4 + j0 + 1];
      v2f a;                         // one-hot A[m=k][p]
      a.x = (sa == row) ? 1.f : 0.f;
      a.y = (sb == row) ? 1.f : 0.f;
      v2f bLo, bHi;                  // B[p][d] rows: lane n = d index
      bLo.x = myE[(g * 4 + j0) * 36 + row];
      bLo.y = myE[(g * 4 + j0 + 1) * 36 + row];
      bHi.x = myE[(g * 4 + j0) * 36 + 16 + row];
      bHi.y = myE[(g * 4 + j0 + 1) * 36 + 16 + row];
      accLo = __builtin_amdgcn_wmma_f32_16x16x4_f32(false, a, false, bLo, (short)0, accLo,
                                                    false, false);
      accHi = __builtin_amdgcn_wmma_f32_16x16x4_f32(false, a, false, bHi, (short)0, accHi,
                                                    false, false);
    }
    __syncthreads();
  }
  // wave accumulators -> block LDS (C/D layout: lane n=d, VGPR r -> k=r (+8 for upper half-wave))
#pragma unroll
  for (int r = 0; r < 8; ++r) {
    const int k = r + ((l < 16) ? 0 : 8);
    atomicAdd(&s_sum[k * 33 + row],      accLo[r]);
    atomicAdd(&s_sum[k * 33 + 16 + row], accHi[r]);
  }
  __syncthreads();
  if (tid < KK) atomicAdd(&ws[OFF_COUNTS + b * KK + tid], s_cnt[tid]);
  for (int i = tid; i < KK * DD; i += 256) {
    const int k = i / DD, d = i % DD;
    atomicAdd(&ws[OFF_SUMS + (b * KK + k) * DD + d], s_sum[k * 33 + d]);
  }
}

// -------- Centers + push/reg per image: one wave32, WMMA f32 Gram matrix --------
__global__ __launch_bounds__(32) void k_centers(float* __restrict__ ws) {
  __shared__ float sC[KK * 33];
  __shared__ float sG[KK * 17];
  __shared__ float sPush[KK], sPairs[KK], sCn2[KK], sValid[KK], sCnorm[KK];
  const int l = threadIdx.x;
  const int b = blockIdx.x;

  if (l < KK) {
    const float cnt = ws[OFF_COUNTS + b * KK + l];
    const float inv = 1.f / fmaxf(cnt, 1.f);
    float cn2 = 0.f;
    for (int d = 0; d < DD; ++d) {
      const float c = ws[OFF_SUMS + (b * KK + l) * DD + d] * inv;
      sC[l * 33 + d] = c;
      ws[OFF_CENTERS + (b * KK + l) * DD + d] = c;
      cn2 += c * c;
    }
    sCn2[l] = cn2;
    sValid[l] = (cnt > 0.f) ? 1.f : 0.f;
    ws[OFF_CNORM2 + b * KK + l] = cn2;
    ws[OFF_VALID + b * KK + l] = sValid[l];
  }
  __syncthreads();

  const int row = l & 15;
  v8f acc = {};
#pragma unroll
  for (int c = 0; c < 8; ++c) {
    v2f a;  // f32 16x4 A layout: lanes 0-15 -> K=4c,4c+1 ; lanes 16-31 -> K=4c+2,4c+3
    if (l < 16) { a.x = sC[row * 33 + 4 * c + 0]; a.y = sC[row * 33 + 4 * c + 1]; }
    else        { a.x = sC[row * 33 + 4 * c + 2]; a.y = sC[row * 33 + 4 * c + 3]; }
    // Gram matrix G = C * C^T : A and B register images are identical for f32 layouts
    acc = __builtin_amdgcn_wmma_f32_16x16x4_f32(false, a, false, a, (short)0, acc, false, false);
  }
#pragma unroll
  for (int r = 0; r < 8; ++r) {
    const int m = r + ((l < 16) ? 0 : 8);   // C/D layout: lanes16-31 hold M=r+8
    sG[m * 17 + row] = acc[r];
  }
  __syncthreads();

  if (l < KK) {
    float pushrow = 0.f, pairs = 0.f;
    const float vmL = sValid[l];
    for (int k = 0; k < KK; ++k) {
      if (k == l) continue;
      const float pm = vmL * sValid[k];
      const float d2 = fmaxf(sCn2[l] + sCn2[k] - 2.f * sG[l * 17 + k], 0.f);
      const float dm = sqrtf(d2);
      const float t  = fmaxf(2.f * DDIST - dm, 0.f);
      pushrow += pm * t * t;
      pairs   += pm;
    }
    sPush[l]  = pushrow;
    sPairs[l] = pairs;
    sCnorm[l] = sValid[l] * sqrtf(sCn2[l]);
  }
  __syncthreads();

  if (l == 0) {
    float pushsum = 0.f, pairsum = 0.f, nids = 0.f, regsum = 0.f, tot = 0.f;
    for (int k = 0; k < KK; ++k) {
      pushsum += sPush[k]; pairsum += sPairs[k];
      nids += sValid[k];   regsum += sCnorm[k];
      tot += ws[OFF_COUNTS + b * KK + k];
    }
    ws[OFF_IMG + b * 4 + 0] = (nids > 1.f) ? pushsum / fmaxf(pairsum, 1.f) : 0.f;
    ws[OFF_IMG + b * 4 + 1] = regsum / fmaxf(nids, 1.f);
    ws[OFF_IMG + b * 4 + 2] = nids;
    ws[OFF_IMG + b * 4 + 3] = (tot > 0.f) ? 1.f : 0.f;
  }
}

// -------- Pass 2: pull term. dots[pix][k] via WMMA f32 16x16x4, 8 chunks of D --------
__global__ __launch_bounds__(256) void k_pull(const float* __restrict__ emb,
                                              const int* __restrict__ inst,
                                              const unsigned char* __restrict__ vmask,
                                              float* __restrict__ ws) {
  __shared__ float sDots[8 * 16 * 17];   // per-wave 16x16 tile, stride 17
  __shared__ float sPull[KK];
  __shared__ float sCtr[KK * DD];
  __shared__ float sCn2v[KK];
  const int tid = threadIdx.x;
  const int b   = blockIdx.x / BLOCKS_PER_IMG;
  const int blk = blockIdx.x % BLOCKS_PER_IMG;

  if (tid < KK) sPull[tid] = 0.f;
  for (int i = tid; i < KK * DD; i += 256) sCtr[i] = ws[OFF_CENTERS + b * KK * DD + i];
  if (tid < KK) sCn2v[tid] = ws[OFF_CNORM2 + b * KK + tid];
  __syncthreads();

  const int wave = tid >> 5;
  const int l    = tid & 31;
  const int row  = l & 15;

  // B regs: B[dim][inst] 4x16 f32 layout -> same lane pattern as A, indexed by instance
  v2f breg[8];
#pragma unroll
  for (int c = 0; c < 8; ++c) {
    if (l < 16) { breg[c].x = sCtr[row * DD + 4 * c + 0]; breg[c].y = sCtr[row * DD + 4 * c + 1]; }
    else        { breg[c].x = sCtr[row * DD + 4 * c + 2]; breg[c].y = sCtr[row * DD + 4 * c + 3]; }
  }

  const float* e0 = emb + (size_t)b * DD * NN;
  const int wbase = blk * PIX_PER_BLOCK + wave * 512;   // 512 pixels per wave
  float* myDots = &sDots[wave * 272];

  for (int t = 0; t < 32; ++t) {        // 32 tiles of 16 pixels
    const int p0 = wbase + t * 16;
    float e2 = 0.f;
    v8f acc = {};
#pragma unroll
    for (int c = 0; c < 8; ++c) {
      const int d0 = (l < 16) ? (4 * c) : (4 * c + 2);
      v2f a;
      a.x = e0[(size_t)d0 * NN + p0 + row];
      a.y = e0[(size_t)(d0 + 1) * NN + p0 + row];
      e2 += a.x * a.x + a.y * a.y;
      acc = __builtin_amdgcn_wmma_f32_16x16x4_f32(false, a, false, breg[c], (short)0, acc,
                                                  false, false);
    }
    __builtin_prefetch(&e0[(size_t)((l < 16) ? 0 : 2) * NN + p0 + 64 + row], 0, 0);
    e2 += __shfl_xor(e2, 16, 32);       // combine the two half-wave dim partials -> ||e||^2

#pragma unroll
    for (int r = 0; r < 8; ++r) {
      const int m = r + ((l < 16) ? 0 : 8);
      myDots[m * 17 + row] = acc[r];
    }
    __syncthreads();

    if (l < 16) {
      const int p = p0 + l;
      const int s = inst[(size_t)b * NN + p];
      const bool msk = vmask[(size_t)b * NN + p] && (s != IGN) && (s >= 0) && (s < KK);
      if (msk) {
        const float dot  = myDots[l * 17 + s];
        const float d2   = fmaxf(e2 - 2.f * dot + sCn2v[s], 0.f);
        const float dist = sqrtf(d2);
        const float v    = fmaxf(dist - DVAR, 0.f);
        atomicAdd(&sPull[s], v * v);
      }
    }
    __syncthreads();
  }
  if (tid < KK) atomicAdd(&ws[OFF_PULL + b * KK + tid], sPull[tid]);
}

// ---------------- Final combine -> scalar loss ----------------
__global__ void k_final(const float* __restrict__ ws, float* __restrict__ out) {
  if (threadIdx.x != 0 || blockIdx.x != 0) return;
  float num = 0.f, nvb = 0.f;
  for (int b = 0; b < BB; ++b) {
    const float nids = ws[OFF_IMG + b * 4 + 2];
    const float vb   = ws[OFF_IMG + b * 4 + 3];
    float pull = 0.f;
    for (int k = 0; k < KK; ++k) {
      const float cnt   = ws[OFF_COUNTS + b * KK + k];
      const float valid = ws[OFF_VALID + b * KK + k];
      pull += valid * ws[OFF_PULL + b * KK + k] / fmaxf(cnt, 1.f);
    }
    pull /= fmaxf(nids, 1.f);
    const float push = ws[OFF_IMG + b * 4 + 0];
    const float reg  = ws[OFF_IMG + b * 4 + 1];
    num += vb * (PULLW * pull + PUSHW * push + REGW * reg);
    nvb += vb;
  }
  out[0] = (nvb > 0.f) ? num / fmaxf(nvb, 1.f) : 0.f;
}

extern "C" void kernel_launch(void* const* d_in, const int* in_sizes, int n_in,
                              void* d_out, int out_size, void* d_ws, size_t ws_size,
                              hipStream_t stream) {
  const float* emb = (const float*)d_in[0];
  const int* inst  = (const int*)d_in[1];
  const unsigned char* vmask = (const unsigned char*)d_in[2];
  float* ws  = (float*)d_ws;
  float* out = (float*)d_out;

  k_zero<<<(WS_FLOATS + 255) / 256, 256, 0, stream>>>(ws);
  k_pass1<<<BB * BLOCKS_PER_IMG, 256, 0, stream>>>(emb, inst, vmask, ws);
  k_centers<<<BB, 32, 0, stream>>>(ws);
  k_pull<<<BB * BLOCKS_PER_IMG, 256, 0, stream>>>(emb, inst, vmask, ws);
  k_final<<<1, 32, 0, stream>>>(ws, out);
}